// ScaledDotProductAttention_19756849562280
// MI455X (gfx1250) — compile-verified
//
#include <hip/hip_runtime.h>

typedef __attribute__((ext_vector_type(2))) float v2f;
typedef __attribute__((ext_vector_type(8))) float v8f;

#define BB 2
#define HH 16
#define SS 2048
#define DD 128
#define QT 16
#define SQ_STRIDE 132   // 128 + 4 pad -> banks (4*row + col) % 64, conflict-free column reads
#define SS_STRIDE 2052  // 2048 + 4 pad, even (keeps float2 LDS loads 8B-aligned)

__global__ __launch_bounds__(256) void ScaledDotProductAttention_19756849562280_kernel(
    const float* __restrict__ q, const float* __restrict__ k,
    const float* __restrict__ v, const int* __restrict__ mask,
    const float* __restrict__ bias, float* __restrict__ out,
    float* __restrict__ attn)
{
  extern __shared__ float smem[];
  float* sQ   = smem;                      // QT * SQ_STRIDE
  float* sS   = sQ + QT * SQ_STRIDE;       // QT * SS_STRIDE
  float* sMax = sS + QT * SS_STRIDE;       // 16*16
  float* sSum = sMax + 256;                // 16*16

  const int tid   = threadIdx.x;
  const int wave  = tid >> 5;
  const int lane  = tid & 31;
  const int lhalf = lane >> 4;   // 0 or 1
  const int l16   = lane & 15;

  const int blk = blockIdx.x;
  const int qt  = blk % (SS / QT);
  const int bh  = blk / (SS / QT);      // b*H + h
  const int h   = bh % HH;
  const int b   = bh / HH;
  const int q0  = qt * QT;

  const float scale = 0.088388347648318447f; // 1/sqrt(128)

  const float* qBase    = q    + ((size_t)bh * SS + q0) * DD;
  const float* kBase    = k    + (size_t)bh * SS * DD;
  const float* vBase    = v    + (size_t)bh * SS * DD;
  const float* biasBase = bias + (size_t)h * SS * SS + (size_t)q0 * SS;
  const int*   maskBase = mask + (size_t)b * SS * SS + (size_t)q0 * SS;

  // ---- load Q tile into LDS (16x128, padded rows) ----
  for (int i = tid; i < QT * DD; i += 256) {
    int r = i >> 7, c = i & (DD - 1);
    sQ[r * SQ_STRIDE + c] = qBase[i];
  }
  __syncthreads();

  // ---- scores = Q @ K^T (each wave: 16 column tiles of 16) ----
  for (int nt = 0; nt < 16; ++nt) {
    const int j0 = (wave * 16 + nt) * 16;
    v8f acc = {};
    // B fragment source: K row (j0 + l16), consecutive d pairs (8B aligned)
    const float* kRow = kBase + (size_t)(j0 + l16) * DD + 2 * lhalf;
    // A fragment source: sQ row l16 (8B aligned, padded stride)
    const float* qRow = sQ + l16 * SQ_STRIDE + 2 * lhalf;
#pragma unroll 4
    for (int d0 = 0; d0 < DD; d0 += 4) {
      v2f a  = *(const v2f*)(qRow + d0);
      v2f bf = *(const v2f*)(kRow + d0);
      acc = __builtin_amdgcn_wmma_f32_16x16x4_f32(false, a, false, bf,
                                                  (short)0, acc, false, false);
    }
    // C/D layout: VGPR j -> row (j + 8*lhalf), lane%16 -> col
#pragma unroll
    for (int j = 0; j < 8; ++j) {
      const int m   = j + 8 * lhalf;
      const int col = j0 + l16;
      float val = acc[j] * scale + biasBase[(size_t)m * SS + col];
      if (maskBase[(size_t)m * SS + col] == 0) val = -1.0e9f;
      sS[m * SS_STRIDE + col] = val;
    }
  }
  __syncthreads();

  // ---- softmax over each of the 16 rows (16 threads per row) ----
  {
    const int row = tid >> 4;
    const int c0  = tid & 15;
    float* rowS = sS + row * SS_STRIDE;

    float m = -3.4e38f;
    for (int i = 0; i < SS / 16; ++i) m = fmaxf(m, rowS[c0 + 16 * i]);
    sMax[row * 16 + c0] = m;
    __syncthreads();

    float rowmax = sMax[row * 16];
#pragma unroll
    for (int i = 1; i < 16; ++i) rowmax = fmaxf(rowmax, sMax[row * 16 + i]);

    float psum = 0.0f;
    for (int i = 0; i < SS / 16; ++i) {
      float e = __expf(rowS[c0 + 16 * i] - rowmax);
      rowS[c0 + 16 * i] = e;
      psum += e;
    }
    sSum[row * 16 + c0] = psum;
    __syncthreads();

    float tot = 0.0f;
#pragma unroll
    for (int i = 0; i < 16; ++i) tot += sSum[row * 16 + i];
    const float inv = 1.0f / tot;

    // attn is a pure streaming output (never re-read by this kernel):
    // non-temporal stores keep the 192MB L2 free for bias/mask/K/V reuse.
    float* attnRow = attn + ((size_t)bh * SS + q0 + row) * SS;
    for (int i = 0; i < SS / 16; ++i) {
      float p = rowS[c0 + 16 * i] * inv;
      rowS[c0 + 16 * i] = p;                          // keep normalized probs for PV
      __builtin_nontemporal_store(p, attnRow + c0 + 16 * i); // coalesced 64B segments
    }
  }
  __syncthreads();

  // ---- out = attn @ V (each wave: 16 output columns) ----
  {
    const int n0 = wave * 16;
    v8f acc = {};
    const float* vCol = vBase + n0 + l16;
    const float* aRow = sS + l16 * SS_STRIDE + 2 * lhalf;
#pragma unroll 4
    for (int kk = 0; kk < SS; kk += 4) {
      v2f a = *(const v2f*)(aRow + kk);
      const int krow = kk + 2 * lhalf;
      v2f bf;
      bf.x = vCol[(size_t)krow * DD];
      bf.y = vCol[(size_t)(krow + 1) * DD];
      acc = __builtin_amdgcn_wmma_f32_16x16x4_f32(false, a, false, bf,
                                                  (short)0, acc, false, false);
    }
    float* outBase = out + ((size_t)bh * SS + q0) * DD + n0;
#pragma unroll
    for (int j = 0; j < 8; ++j) {
      const int m = j + 8 * lhalf;
      __builtin_nontemporal_store(acc[j], outBase + (size_t)m * DD + l16);
    }
  }
}

extern "C" void kernel_launch(void* const* d_in, const int* in_sizes, int n_in,
                              void* d_out, int out_size, void* d_ws, size_t ws_size,
                              hipStream_t stream) {
  (void)in_sizes; (void)n_in; (void)d_ws; (void)ws_size; (void)out_size;
  const float* q    = (const float*)d_in[0];
  const float* k    = (const float*)d_in[1];
  const float* v    = (const float*)d_in[2];
  const int*   mask = (const int*)d_in[3];
  const float* bias = (const float*)d_in[4];

  float* out  = (float*)d_out;                               // [B,H,S,D]
  float* attn = out + (size_t)BB * HH * SS * DD;             // [B,H,S,S]

  const size_t smemBytes =
      (size_t)(QT * SQ_STRIDE + QT * SS_STRIDE + 256 + 256) * sizeof(float);
  const int grid = BB * HH * (SS / QT); // 4096 blocks, one per (b,h,16-row q tile)

  ScaledDotProductAttention_19756849562280_kernel<<<grid, 256, smemBytes, stream>>>(
      q, k, v, mask, bias, out, attn);
}